// GRN_73684458930478
// MI455X (gfx1250) — compile-verified
//
#include <hip/hip_runtime.h>
#include <hip/hip_bf16.h>
#include <math.h>

// ---------------------------------------------------------------------------
// Problem constants (from reference): B=16, S=512, N=16, WD=300, ED=64,
// SD=512, V=100, L=3.  M = B*S = 8192.  K1 = WD+ED = 364, K2 = SD+ED = 576.
// ---------------------------------------------------------------------------
#define MROWS   8192
#define NNBR    16
#define WD_     300
#define ED_     64
#define SD_     512
#define K1_     364      // WD+ED
#define K2_     576      // SD+ED
#define NGATE4  2048     // 4 gates * 512
#define NLAYERS 3

typedef __attribute__((ext_vector_type(2))) float v2f;
typedef __attribute__((ext_vector_type(8))) float v8f;

// ---------------------------------------------------------------------------
// Kernel 1: masked neighbor gather-sum (memory-bound; ~0.5 GB reads/layer).
//   out[node, 0:D]     = sum_n mask[node,n] * rep[batch(node), idx[node,n], 0:D]
//   out[node, D:D+64]  = sum_n mask[node,n] * edge_emb[edges[node,n], 0:64]
//   optionally scaled by seq_mask[node]  (pass nullptr to skip)
// ---------------------------------------------------------------------------
__global__ void __launch_bounds__(256)
gather_sum_kernel(const float* __restrict__ rep, int D,
                  const float* __restrict__ edge_emb,
                  const int*   __restrict__ idx,
                  const int*   __restrict__ edges,
                  const float* __restrict__ mask,
                  const float* __restrict__ seq_mask,
                  float*       __restrict__ outp)
{
    __shared__ int   s_idx[NNBR];
    __shared__ int   s_edg[NNBR];
    __shared__ float s_msk[NNBR];

    const int node = blockIdx.x;            // 0..MROWS-1
    const int base = (node >> 9) << 9;      // batch * S (S = 512)

    if (threadIdx.x < NNBR) {
        s_idx[threadIdx.x] = idx  [node * NNBR + threadIdx.x];
        s_edg[threadIdx.x] = edges[node * NNBR + threadIdx.x];
        s_msk[threadIdx.x] = mask [node * NNBR + threadIdx.x];
    }
    __syncthreads();

    const float sm   = seq_mask ? seq_mask[node] : 1.0f;
    const int   Wtot = D + ED_;

    for (int f = threadIdx.x; f < Wtot; f += 256) {
        float acc = 0.0f;
        if (f < D) {
            #pragma unroll
            for (int n = 0; n < NNBR; ++n)
                acc += s_msk[n] * rep[(size_t)(base + s_idx[n]) * D + f];
        } else {
            const int fe = f - D;
            #pragma unroll
            for (int n = 0; n < NNBR; ++n)
                acc += s_msk[n] * edge_emb[s_edg[n] * ED_ + fe];
        }
        outp[(size_t)node * Wtot + f] = acc * sm;
    }
}

// ---------------------------------------------------------------------------
// Kernel 2: w_terms GEMM (fp32 WMMA 16x16x4).
//   C[M,2048] = Ain[M,364] @ Win'[364,2048] + Aout[M,364] @ Wout'[364,2048] + b
// Weight storage [4,364,512]; column col -> (g = col>>9, d = col&511),
// element (k,col) at W[(g*364 + k)*512 + d].
// Block: 256 threads = 8 waves, all tiling N; wave tile 32 rows x 64 cols
// (acc[2][4]) so each B fragment feeds two WMMAs (1.5 loads/WMMA).
// Grid: (M/32, 2048/512) = (256, 4).
// ---------------------------------------------------------------------------
__global__ void __launch_bounds__(256)
wterms_gemm(const float* __restrict__ Ain,
            const float* __restrict__ Aout,
            const float* __restrict__ Win,
            const float* __restrict__ Wout,
            const float* __restrict__ bias,
            float*       __restrict__ Cmat)
{
    const int lane = threadIdx.x & 31;
    const int wn   = threadIdx.x >> 5;   // 0..7 : N sub-tile
    const int half = lane >> 4;          // 0 or 1
    const int r    = lane & 15;
    const int m0   = blockIdx.x * 32;
    const int n0   = blockIdx.y * 512 + wn * 64;

    v8f acc[2][4] = {};

    for (int src = 0; src < 2; ++src) {
        const float* __restrict__ A = src ? Aout : Ain;
        const float* __restrict__ W = src ? Wout : Win;
        const float* __restrict__ arow0 = A + (size_t)(m0 + r) * K1_;
        const float* __restrict__ arow1 = A + (size_t)(m0 + 16 + r) * K1_;
        for (int k = 0; k < K1_; k += 4) {
            v2f a0, a1;
            a0.x = arow0[k + 2 * half];
            a0.y = arow0[k + 2 * half + 1];
            a1.x = arow1[k + 2 * half];
            a1.y = arow1[k + 2 * half + 1];
            #pragma unroll
            for (int t = 0; t < 4; ++t) {
                const int col = n0 + t * 16 + r;
                const int g   = col >> 9;
                const int d   = col & 511;
                const float* __restrict__ wp =
                    W + ((size_t)(g * K1_ + k + 2 * half)) * SD_ + d;
                v2f bfr;
                bfr.x = wp[0];
                bfr.y = wp[SD_];
                acc[0][t] = __builtin_amdgcn_wmma_f32_16x16x4_f32(
                    false, a0, false, bfr, (short)0, acc[0][t], false, false);
                acc[1][t] = __builtin_amdgcn_wmma_f32_16x16x4_f32(
                    false, a1, false, bfr, (short)0, acc[1][t], false, false);
            }
        }
    }

    #pragma unroll
    for (int mi = 0; mi < 2; ++mi) {
        #pragma unroll
        for (int t = 0; t < 4; ++t) {
            const int   col = n0 + t * 16 + r;
            const float bv  = bias[col];
            #pragma unroll
            for (int v = 0; v < 8; ++v) {
                const int row = m0 + mi * 16 + v + 8 * half;
                Cmat[(size_t)row * NGATE4 + col] = acc[mi][t][v] + bv;
            }
        }
    }
}

// ---------------------------------------------------------------------------
// Kernel 3: fused per-layer GEMM + LSTM.
//   g[gate] = w_terms[:, gate*512+d] + in_h @ Ug_in[gate] + out_h @ Ug_out[gate]
//   then the LSTM cell/hidden update, scaled by seq_mask.
// Block: 8 waves = 4(gate) x 2(d-subtile); wave tile 32 rows x 64 d-cols
// (acc[2][4], B reuse across the two M sub-tiles).
// Block tile: 32 rows x 128 d-cols x all 4 gates -> LSTM fused via a 64 KB
// LDS exchange (column XOR-swizzled by the C-fragment half to stay
// bank-conflict free with a pitch that is a multiple of 64).
// Grid: (M/32, 512/128) = (256, 4).
// ---------------------------------------------------------------------------
#define LDS_PITCH 128

__device__ __forceinline__ float sigmoidf_(float x)
{
    return 1.0f / (1.0f + __expf(-x));
}

__global__ void __launch_bounds__(256)
layer_gemm_lstm(const float* __restrict__ in_h,     // [M, 576]
                const float* __restrict__ out_h,    // [M, 576]
                const float* __restrict__ Uin,      // [4, 576, 512]
                const float* __restrict__ Uout,     // [4, 576, 512]
                const float* __restrict__ wterms,   // [M, 2048]
                const float* __restrict__ seq_mask, // [M]
                float*       __restrict__ cell,     // [M, 512] (read old, write new)
                float*       __restrict__ hid_out,  // reps slice [M, 512]
                float*       __restrict__ hid_out2, // node_hidden output or nullptr
                float*       __restrict__ cell_out2,// node_cell output or nullptr
                int first_layer)
{
    __shared__ float lds[4 * 32 * LDS_PITCH];      // 64 KB

    const int lane = threadIdx.x & 31;
    const int wave = threadIdx.x >> 5;
    const int gate = wave >> 1;          // 0..3: in, forget, out, cellinput
    const int wd   = wave & 1;           // d sub-tile
    const int half = lane >> 4;
    const int r    = lane & 15;
    const int m0   = blockIdx.x * 32;
    const int d0   = blockIdx.y * 128 + wd * 64;   // d column base within [0,512)

    v8f acc[2][4] = {};

    for (int src = 0; src < 2; ++src) {
        const float* __restrict__ H  = src ? out_h : in_h;
        const float* __restrict__ U  = src ? Uout : Uin;
        const float* __restrict__ h0 = H + (size_t)(m0 + r) * K2_;
        const float* __restrict__ h1 = H + (size_t)(m0 + 16 + r) * K2_;
        const float* __restrict__ ug = U + (size_t)gate * K2_ * SD_;
        for (int k = 0; k < K2_; k += 4) {
            v2f a0, a1;
            a0.x = h0[k + 2 * half];
            a0.y = h0[k + 2 * half + 1];
            a1.x = h1[k + 2 * half];
            a1.y = h1[k + 2 * half + 1];
            const float* __restrict__ up =
                ug + (size_t)(k + 2 * half) * SD_ + d0 + r;
            #pragma unroll
            for (int t = 0; t < 4; ++t) {
                v2f bfr;
                bfr.x = up[t * 16];
                bfr.y = up[t * 16 + SD_];
                acc[0][t] = __builtin_amdgcn_wmma_f32_16x16x4_f32(
                    false, a0, false, bfr, (short)0, acc[0][t], false, false);
                acc[1][t] = __builtin_amdgcn_wmma_f32_16x16x4_f32(
                    false, a1, false, bfr, (short)0, acc[1][t], false, false);
            }
        }
    }

    // add loop-invariant w_terms and park gate tiles in LDS (swizzled)
    #pragma unroll
    for (int mi = 0; mi < 2; ++mi) {
        #pragma unroll
        for (int t = 0; t < 4; ++t) {
            const int col = d0 + t * 16 + r;                 // d within [0,512)
            #pragma unroll
            for (int v = 0; v < 8; ++v) {
                const int grow = m0 + mi * 16 + v + 8 * half;
                const int lrow = mi * 16 + v + 8 * half;     // 0..31
                const int lc   = (wd * 64 + t * 16 + r) ^ (half << 4);
                const float gval =
                    acc[mi][t][v] + wterms[(size_t)grow * NGATE4 + gate * SD_ + col];
                lds[(gate * 32 + lrow) * LDS_PITCH + lc] = gval;
            }
        }
    }
    __syncthreads();

    // fused LSTM elementwise over the 32 x 128 block tile
    #pragma unroll
    for (int i = 0; i < 16; ++i) {
        const int e    = threadIdx.x + i * 256;   // 0..4095
        const int row  = e >> 7;                  // 0..31
        const int col  = e & 127;
        const int pcol = col ^ (((row >> 3) & 1) << 4);  // undo swizzle
        const int grow = blockIdx.x * 32 + row;
        const int gcol = blockIdx.y * 128 + col;

        const float ig = sigmoidf_(lds[(0 * 32 + row) * LDS_PITCH + pcol]);
        const float fg = sigmoidf_(lds[(1 * 32 + row) * LDS_PITCH + pcol]);
        const float og = sigmoidf_(lds[(2 * 32 + row) * LDS_PITCH + pcol]);
        const float ci = tanhf     (lds[(3 * 32 + row) * LDS_PITCH + pcol]);

        const float cold = first_layer ? 0.0f : cell[(size_t)grow * SD_ + gcol];
        const float sm   = seq_mask[grow];

        float c = fg * cold + ig * ci;
        float h = og * tanhf(c);
        c *= sm;
        h *= sm;

        cell[(size_t)grow * SD_ + gcol]    = c;
        hid_out[(size_t)grow * SD_ + gcol] = h;
        if (hid_out2)  hid_out2 [(size_t)grow * SD_ + gcol] = h;
        if (cell_out2) cell_out2[(size_t)grow * SD_ + gcol] = c;
    }
}

// ---------------------------------------------------------------------------
// Host-side orchestration.
// ---------------------------------------------------------------------------
extern "C" void kernel_launch(void* const* d_in, const int* in_sizes, int n_in,
                              void* d_out, int out_size, void* d_ws, size_t ws_size,
                              hipStream_t stream)
{
    const float* word_repres     = (const float*)d_in[0];   // [16,512,300]
    const float* sentence_repres = (const float*)d_in[1];   // [16,512,512]
    const float* seq_mask        = (const float*)d_in[2];   // [16,512]
    const float* in_mask         = (const float*)d_in[3];   // [16,512,16]
    const float* out_mask        = (const float*)d_in[4];   // [16,512,16]
    const float* edge_emb        = (const float*)d_in[5];   // [100,64]
    const float* Wg_in           = (const float*)d_in[6];   // [4,364,512]
    const float* Ug_in           = (const float*)d_in[7];   // [4,576,512]
    const float* Wg_out          = (const float*)d_in[8];   // [4,364,512]
    const float* Ug_out          = (const float*)d_in[9];   // [4,576,512]
    const float* bias            = (const float*)d_in[10];  // [4,512] == [2048]
    const int*   in_idx          = (const int*)d_in[11];
    const int*   in_edges        = (const int*)d_in[12];
    const int*   out_idx         = (const int*)d_in[13];
    const int*   out_edges       = (const int*)d_in[14];
    // d_in[15] = num_layers (device scalar; L=3 per reference constants)

    float* out = (float*)d_out;   // [3*M*512 reps][M*512 hidden][M*512 cell]

    // Workspace carve-up (floats): in/out buffers sized for the wider K2 phase.
    float* ws      = (float*)d_ws;
    float* in_buf  = ws;                                 // M*576 (also holds in_rep  M*364)
    float* out_buf = in_buf + (size_t)MROWS * K2_;       // M*576 (also holds out_rep M*364)
    float* wterms  = out_buf + (size_t)MROWS * K2_;      // M*2048
    float* cellbuf = wterms + (size_t)MROWS * NGATE4;    // M*512

    const dim3 gatherGrid(MROWS);
    const dim3 gemmGrid1(MROWS / 32, NGATE4 / 512);      // (256, 4)
    const dim3 layerGrid(MROWS / 32, SD_ / 128);         // (256, 4)

    // ---- static neighbor word+edge sums (no seq-mask scaling) ----
    gather_sum_kernel<<<gatherGrid, 256, 0, stream>>>(
        word_repres, WD_, edge_emb, in_idx, in_edges, in_mask, nullptr, in_buf);
    gather_sum_kernel<<<gatherGrid, 256, 0, stream>>>(
        word_repres, WD_, edge_emb, out_idx, out_edges, out_mask, nullptr, out_buf);

    // ---- loop-invariant w_terms for all 4 gates ----
    wterms_gemm<<<gemmGrid1, 256, 0, stream>>>(
        in_buf, out_buf, Wg_in, Wg_out, bias, wterms);

    // ---- recurrent layers ----
    for (int l = 0; l < NLAYERS; ++l) {
        const float* hidden = (l == 0) ? sentence_repres
                                       : (out + (size_t)(l - 1) * MROWS * SD_);

        gather_sum_kernel<<<gatherGrid, 256, 0, stream>>>(
            hidden, SD_, edge_emb, in_idx, in_edges, in_mask, seq_mask, in_buf);
        gather_sum_kernel<<<gatherGrid, 256, 0, stream>>>(
            hidden, SD_, edge_emb, out_idx, out_edges, out_mask, seq_mask, out_buf);

        float* hid_out   = out + (size_t)l * MROWS * SD_;
        const bool last  = (l == NLAYERS - 1);
        float* hid_out2  = last ? (out + (size_t)NLAYERS * MROWS * SD_) : nullptr;
        float* cell_out2 = last ? (out + (size_t)(NLAYERS + 1) * MROWS * SD_) : nullptr;

        layer_gemm_lstm<<<layerGrid, 256, 0, stream>>>(
            in_buf, out_buf, Ug_in, Ug_out, wterms, seq_mask,
            cellbuf, hid_out, hid_out2, cell_out2, (l == 0) ? 1 : 0);
    }
}